// VectorQuantizer_62474594288297
// MI455X (gfx1250) — compile-verified
//
#include <hip/hip_runtime.h>
#include <hip/hip_bf16.h>
#include <math.h>

typedef __attribute__((ext_vector_type(2))) float v2f;
typedef __attribute__((ext_vector_type(8))) float v8f;

#define N_CODES 1024
#define E_DIM   64
#define BETA    0.25f

// ---------------------------------------------------------------------------
// Kernel 0: zero accumulators, precompute codebook squared norms.
// ---------------------------------------------------------------------------
__global__ void __launch_bounds__(128)
vq_init_kernel(const float* __restrict__ cb, float* __restrict__ cnorm,
               float* __restrict__ counts, float* __restrict__ sse) {
  int i = blockIdx.x * blockDim.x + threadIdx.x;
  if (i == 0) *sse = 0.0f;
  if (i < N_CODES) {
    counts[i] = 0.0f;
    const float* row = cb + i * E_DIM;
    float s = 0.0f;
#pragma unroll
    for (int k = 0; k < E_DIM; ++k) s = fmaf(row[k], row[k], s);
    cnorm[i] = s;
  }
}

// ---------------------------------------------------------------------------
// Kernel 1: argmin over codes via fp32 WMMA (v_wmma_f32_16x16x4_f32).
// One wave owns TWO 16-token M-tiles (32 tokens); loops over 64 code N-tiles.
// Each B fragment feeds two WMMAs (c0/c1 interleaved -> RAW distance 2), and
// the next tile's B fragments are prefetched (register double buffer) so
// s_wait_loadcnt does not sit between dependent WMMAs.
// dist = ||e||^2 - 2*dot (||z||^2 is constant per row -> dropped).
// ---------------------------------------------------------------------------
__global__ void __launch_bounds__(128)
vq_argmin_kernel(const float* __restrict__ z, const float* __restrict__ cb,
                 const float* __restrict__ cnorm, int* __restrict__ idx_out) {
  const int lane = threadIdx.x & 31;
  const int wave = threadIdx.x >> 5;
  const int tm   = (blockIdx.x * 4 + wave) * 2;   // two consecutive M-tiles
  const int half = lane >> 4;                     // 0: lanes 0-15, 1: 16-31
  const int lm   = lane & 15;

  // A fragments per ISA layout: lane m = lm, VGPR0 = A[m, 4s+2h], VGPR1 = +1.
  // z is streamed exactly once -> non-temporal.
  const float* zrow0 = z + (tm * 16 + lm) * E_DIM + 2 * half;
  const float* zrow1 = zrow0 + 16 * E_DIM;
  v2f a0[16], a1[16];
#pragma unroll
  for (int s = 0; s < 16; ++s) {
    a0[s] = __builtin_nontemporal_load((const v2f*)(zrow0 + 4 * s));
    a1[s] = __builtin_nontemporal_load((const v2f*)(zrow1 + 4 * s));
  }

  float minv0[8], minv1[8];
  int   mini0[8], mini1[8];
#pragma unroll
  for (int r = 0; r < 8; ++r) {
    minv0[r] = 3.402823466e38f; mini0[r] = 0;
    minv1[r] = 3.402823466e38f; mini1[r] = 0;
  }

  // Preload B fragments for tileN = 0 (codebook stays cache-resident: RT).
  v2f bc[16];
  {
    const float* crow = cb + lm * E_DIM + 2 * half;
#pragma unroll
    for (int s = 0; s < 16; ++s) bc[s] = *(const v2f*)(crow + 4 * s);
  }

  for (int tileN = 0; tileN < N_CODES / 16; ++tileN) {
    const int code = tileN * 16 + lm;             // B lane = N column

    // Prefetch next tile's B fragments (last iteration reloads same tile).
    const int nn = (tileN + 1 < N_CODES / 16) ? tileN + 1 : tileN;
    const float* crown = cb + (nn * 16 + lm) * E_DIM + 2 * half;
    v2f bn[16];
#pragma unroll
    for (int s = 0; s < 16; ++s) bn[s] = *(const v2f*)(crown + 4 * s);

    v8f c0 = {}, c1 = {};
#pragma unroll
    for (int s = 0; s < 16; ++s) {
      c0 = __builtin_amdgcn_wmma_f32_16x16x4_f32(false, a0[s], false, bc[s],
                                                 (short)0, c0, false, false);
      c1 = __builtin_amdgcn_wmma_f32_16x16x4_f32(false, a1[s], false, bc[s],
                                                 (short)0, c1, false, false);
    }

    const float cn = cnorm[code];
#pragma unroll
    for (int r = 0; r < 8; ++r) {                 // VGPR r -> row m = r+8*half
      float d0 = fmaf(-2.0f, c0[r], cn);
      if (d0 < minv0[r]) { minv0[r] = d0; mini0[r] = code; }
      float d1 = fmaf(-2.0f, c1[r], cn);
      if (d1 < minv1[r]) { minv1[r] = d1; mini1[r] = code; }
    }

#pragma unroll
    for (int s = 0; s < 16; ++s) bc[s] = bn[s];
  }

  // argmin across the 16 lanes of each half; tie-break to lowest index
  // to match jnp.argmin (first-minimum) semantics.
#pragma unroll
  for (int off = 1; off < 16; off <<= 1) {
#pragma unroll
    for (int r = 0; r < 8; ++r) {
      float ov0 = __shfl_xor(minv0[r], off, 32);
      int   oi0 = __shfl_xor(mini0[r], off, 32);
      if (ov0 < minv0[r] || (ov0 == minv0[r] && oi0 < mini0[r])) {
        minv0[r] = ov0; mini0[r] = oi0;
      }
      float ov1 = __shfl_xor(minv1[r], off, 32);
      int   oi1 = __shfl_xor(mini1[r], off, 32);
      if (ov1 < minv1[r] || (ov1 == minv1[r] && oi1 < mini1[r])) {
        minv1[r] = ov1; mini1[r] = oi1;
      }
    }
  }
  if (lm == 0) {
#pragma unroll
    for (int r = 0; r < 8; ++r) {
      idx_out[tm * 16 + r + 8 * half]       = mini0[r];
      idx_out[(tm + 1) * 16 + r + 8 * half] = mini1[r];
    }
  }
}

// ---------------------------------------------------------------------------
// Kernel 2: gather z_q = codebook[idx], accumulate SSE and code histogram.
// One wave per token: lane handles 2 of the 64 dims.
// ---------------------------------------------------------------------------
__global__ void __launch_bounds__(256)
vq_gather_kernel(const float* __restrict__ z, const float* __restrict__ cb,
                 const int* __restrict__ idx, float* __restrict__ zq,
                 float* __restrict__ counts, float* __restrict__ sse) {
  const int lane  = threadIdx.x & 31;
  const int wave  = threadIdx.x >> 5;
  const int token = blockIdx.x * 8 + wave;

  const int code = idx[token];
  const float* crow = cb + code * E_DIM;
  const float* zrow = z + token * E_DIM;
  float* qrow = zq + token * E_DIM;

  float local = 0.0f;
#pragma unroll
  for (int j = 0; j < 2; ++j) {
    int k = lane + 32 * j;
    float q = crow[k];                               // cache-resident gather
    float d = q - __builtin_nontemporal_load(&zrow[k]);
    __builtin_nontemporal_store(q, &qrow[k]);        // write-once output
    local = fmaf(d, d, local);
  }
#pragma unroll
  for (int off = 16; off; off >>= 1) local += __shfl_xor(local, off, 32);

  __shared__ float red[8];
  if (lane == 0) red[wave] = local;
  __syncthreads();
  if (threadIdx.x == 0) {
    float s = 0.0f;
#pragma unroll
    for (int w = 0; w < 8; ++w) s += red[w];
    atomicAdd(sse, s);
  }
  if (lane == 0) atomicAdd(&counts[code], 1.0f);
}

// ---------------------------------------------------------------------------
// Kernel 3: loss = (1+beta)*SSE/numel ; perplexity from histogram.
// ---------------------------------------------------------------------------
__global__ void __launch_bounds__(1024)
vq_final_kernel(const float* __restrict__ counts, const float* __restrict__ sse,
                float* __restrict__ loss_out, float* __restrict__ ppl_out,
                float inv_tokens, float inv_elems) {
  const int i    = threadIdx.x;
  const int lane = i & 31;
  const int wv   = i >> 5;

  float e = counts[i] * inv_tokens;
  float t = e * logf(e + 1e-10f);
#pragma unroll
  for (int off = 16; off; off >>= 1) t += __shfl_xor(t, off, 32);

  __shared__ float red[32];
  if (lane == 0) red[wv] = t;
  __syncthreads();
  if (wv == 0) {
    float s = red[lane];
#pragma unroll
    for (int off = 16; off; off >>= 1) s += __shfl_xor(s, off, 32);
    if (lane == 0) *ppl_out = expf(-s);
  }
  if (i == 0) *loss_out = (1.0f + BETA) * (*sse) * inv_elems;
}

// ---------------------------------------------------------------------------
extern "C" void kernel_launch(void* const* d_in, const int* in_sizes, int n_in,
                              void* d_out, int out_size, void* d_ws, size_t ws_size,
                              hipStream_t stream) {
  const float* z  = (const float*)d_in[0];     // [32*4096*64]
  const float* cb = (const float*)d_in[1];     // [1024*64]
  const int n_tokens = in_sizes[0] / E_DIM;    // 131072

  float* out  = (float*)d_out;
  float* loss = out;                           // [0]
  float* zq   = out + 1;                       // [1 .. n_tokens*64]
  float* ppl  = out + 1 + in_sizes[0];         // last element

  // workspace layout (floats): sse[1] | counts[1024] | cnorm[1024] | idx[n_tokens]
  float* sse    = (float*)d_ws;
  float* counts = sse + 1;
  float* cnorm  = counts + N_CODES;
  int*   idx    = (int*)(cnorm + N_CODES);

  vq_init_kernel<<<(N_CODES + 127) / 128, 128, 0, stream>>>(cb, cnorm, counts, sse);
  // 4 waves/block, 32 tokens/wave -> 128 tokens/block
  vq_argmin_kernel<<<n_tokens / 128, 128, 0, stream>>>(z, cb, cnorm, idx);
  // 8 waves/block, 1 token/wave
  vq_gather_kernel<<<n_tokens / 8, 256, 0, stream>>>(z, cb, idx, zq, counts, sse);
  vq_final_kernel<<<1, 1024, 0, stream>>>(counts, sse, loss, ppl,
                                          1.0f / (float)n_tokens,
                                          1.0f / (float)(n_tokens * E_DIM));
}